// MMCL_78950088835454
// MI455X (gfx1250) — compile-verified
//
#include <hip/hip_runtime.h>
#include <hip/hip_bf16.h>

// Problem constants (from reference): B=2048, N=32768, k=int(0.01*(N-1))=327.
// Strategy: 1 workgroup per row. Stage row (128KB) HBM->LDS once via CDNA5
// async-to-LDS (ASYNCcnt), then radix-select the k-th largest value in LDS,
// then logsumexp over {pos} U {top-k negatives}. HBM traffic = 256MB total
// => ~11us floor at 23.3 TB/s.

#define TPB 1024
#define MAX_N 32768

// Builtin signature (per hipcc diagnostic): param0 = int4 addrspace(1)*,
// param1 = int4 addrspace(3)*, then imm offset, imm cpol.
typedef int v4i __attribute__((ext_vector_type(4)));
typedef __attribute__((address_space(1))) v4i gv4i;
typedef __attribute__((address_space(3))) v4i lv4i;

#if defined(__has_builtin)
#if __has_builtin(__builtin_amdgcn_global_load_async_to_lds_b128)
#define HAVE_ASYNC_LDS 1
#endif
#if __has_builtin(__builtin_amdgcn_s_wait_asynccnt)
#define HAVE_WAIT_ASYNC 1
#endif
#endif

__device__ __forceinline__ unsigned fkey(float f) {
  // Monotone map: larger float -> larger unsigned key.
  unsigned u = __float_as_uint(f);
  return (u & 0x80000000u) ? ~u : (u | 0x80000000u);
}
__device__ __forceinline__ float keyf(unsigned k) {
  unsigned u = (k & 0x80000000u) ? (k & 0x7FFFFFFFu) : ~k;
  return __uint_as_float(u);
}

__global__ __launch_bounds__(TPB, 1) void mmcl_topk_ce_kernel(
    const float* __restrict__ logits, const int* __restrict__ targets,
    float* __restrict__ loss_ws, int N, int K) {
  __shared__ float row_s[MAX_N];            // 128 KB staged row
  __shared__ unsigned hist[256];            // radix histogram
  __shared__ unsigned s_prefix, s_want, s_maxkey;
  __shared__ float s_partial[TPB / 32];

  const int b = blockIdx.x;
  const int tid = threadIdx.x;
  const float* g = logits + (size_t)b * (size_t)N;
  const int tgt = targets[b];

  // ---- Stage row into LDS (single HBM read of the row) -------------------
#ifdef HAVE_ASYNC_LDS
  {
    const int vecs = N >> 2;  // 16B transfers
    for (int v = tid; v < vecs; v += TPB) {
      const int e = v << 2;
      __builtin_amdgcn_global_load_async_to_lds_b128(
          (gv4i*)(g + e), (lv4i*)(&row_s[e]), 0, 0);
    }
#ifdef HAVE_WAIT_ASYNC
    __builtin_amdgcn_s_wait_asynccnt(0);
#else
    asm volatile("s_wait_asynccnt 0" ::: "memory");
#endif
  }
#else
  {
    const int vecs = N >> 2;
    for (int v = tid; v < vecs; v += TPB) {
      float4 t = ((const float4*)g)[v];
      ((float4*)row_s)[v] = t;
    }
  }
#endif
  if (tid == 0) s_maxkey = 0u;
  __syncthreads();

  // ---- Radix-select the K-th largest masked value (exact, with ties) -----
  unsigned prefix = 0u;
  unsigned want = (unsigned)K;  // still-needed count at current prefix level
  for (int shift = 24; shift >= 0; shift -= 8) {
    if (tid < 256) hist[tid] = 0u;
    __syncthreads();

    const unsigned himask = (shift == 24) ? 0u : (0xFFFFFFFFu << (shift + 8));
    for (int i = tid; i < N; i += TPB) {
      if (i == tgt) continue;  // positive column is masked out
      const unsigned kk = fkey(row_s[i]);
      if (shift == 24) atomicMax(&s_maxkey, kk);
      if (((kk ^ prefix) & himask) == 0u)
        atomicAdd(&hist[(kk >> shift) & 0xFFu], 1u);
    }
    __syncthreads();

    if (tid == 0) {
      unsigned cum = 0u;
      for (int d = 255; d >= 0; --d) {
        const unsigned c = hist[d];
        if (cum + c >= want) {
          s_prefix = prefix | ((unsigned)d << shift);
          s_want = want - cum;  // needed among elements with this digit
          break;
        }
        cum += c;
      }
    }
    __syncthreads();
    prefix = s_prefix;
    want = s_want;
  }
  const unsigned Tkey = prefix;      // exact K-th largest key
  const unsigned tiecnt = want;      // how many ==T elements top_k includes

  // ---- logsumexp over {pos} U {top-K negatives} --------------------------
  const float pos = row_s[tgt];
  const float maxv = keyf(s_maxkey);          // largest masked value (top-1 neg)
  const float m = 10.0f * fmaxf(pos, maxv);   // stable shift
  const float Tval = keyf(Tkey);

  float lsum = 0.0f;
  for (int i = tid; i < N; i += TPB) {
    if (i == tgt) continue;
    const unsigned kk = fkey(row_s[i]);
    if (kk > Tkey) lsum += expf(10.0f * row_s[i] - m);
  }
  if (tid == 0) {
    lsum += (float)tiecnt * expf(10.0f * Tval - m);  // tied-at-threshold terms
    lsum += expf(10.0f * pos - m);                   // positive term (row[tgt])
  }

  // Block reduction (wave32 shuffles, then cross-wave via LDS).
  for (int o = 16; o > 0; o >>= 1) lsum += __shfl_down(lsum, o, 32);
  const int wave = tid >> 5, lane = tid & 31;
  if (lane == 0) s_partial[wave] = lsum;
  __syncthreads();
  if (tid < 32) {
    float t = s_partial[tid];
    for (int o = 16; o > 0; o >>= 1) t += __shfl_down(t, o, 32);
    if (tid == 0) loss_ws[b] = m + logf(t) - 10.0f * pos;
  }
}

__global__ void mmcl_mean_kernel(const float* __restrict__ ws,
                                 float* __restrict__ out, int B) {
  __shared__ float sh[256];
  float s = 0.0f;
  for (int i = threadIdx.x; i < B; i += 256) s += ws[i];
  sh[threadIdx.x] = s;
  __syncthreads();
  for (int off = 128; off > 0; off >>= 1) {
    if (threadIdx.x < (unsigned)off) sh[threadIdx.x] += sh[threadIdx.x + off];
    __syncthreads();
  }
  if (threadIdx.x == 0) out[0] = sh[0] / (float)B;
}

extern "C" void kernel_launch(void* const* d_in, const int* in_sizes, int n_in,
                              void* d_out, int out_size, void* d_ws,
                              size_t ws_size, hipStream_t stream) {
  const float* logits = (const float*)d_in[0];
  const int* targets = (const int*)d_in[1];
  const int total = in_sizes[0];
  const int B = in_sizes[1];
  const int N = total / B;                      // 32768
  const int K = (int)(0.01 * (double)(N - 1));  // 327, matches int(r*(N-1))

  float* row_loss = (float*)d_ws;               // B floats of scratch

  mmcl_topk_ce_kernel<<<dim3(B), dim3(TPB), 0, stream>>>(logits, targets,
                                                         row_loss, N, K);
  mmcl_mean_kernel<<<dim3(1), dim3(256), 0, stream>>>(row_loss, (float*)d_out,
                                                      B);
}